// SelectiveSSM_57157424775520
// MI455X (gfx1250) — compile-verified
//
#include <hip/hip_runtime.h>
#include <hip/hip_bf16.h>

typedef __attribute__((ext_vector_type(2))) float v2f;
typedef __attribute__((ext_vector_type(8))) float v8f;
typedef int v4i_vs __attribute__((vector_size(16)));   // matches builtin's V4i

#define B_SZ    8
#define SEQ     2048
#define DMODEL  1024
#define DSTATE  16
#define DCONV   4
#define DTRANK  64
#define E_DIM   96                 // DTRANK + 2*DSTATE
#define MROWS   (B_SZ * SEQ)       // 16384 rows for the two GEMMs

#define KC      128                // K-chunk for GEMM1 LDS staging
#define BSTRIDE 132                // padded LDS col stride: 132 % 64 == 4 -> 64 banks hit
#define BSTRIDE2 68                // GEMM2 padded stride: 68 % 64 == 4

// ---------------------------------------------------------------------------
// CDNA5 async global->LDS copy (ASYNCcnt-tracked), with graceful fallbacks.
// Probe-confirmed prototype: (v4i AS1* src, v4i AS3* dst, imm offset, imm cpol)
// ---------------------------------------------------------------------------
#if defined(__has_builtin)
#  if __has_builtin(__builtin_amdgcn_global_load_async_to_lds_b128)
#    define HAVE_ASYNC_LDS 1
#  else
#    define HAVE_ASYNC_LDS 0
#  endif
#  if __has_builtin(__builtin_amdgcn_s_wait_asynccnt)
#    define HAVE_ASYNC_WAIT_BUILTIN 1
#  else
#    define HAVE_ASYNC_WAIT_BUILTIN 0
#  endif
#else
#  define HAVE_ASYNC_LDS 0
#  define HAVE_ASYNC_WAIT_BUILTIN 0
#endif

__device__ __forceinline__ void async_copy16(const float* g, float* l) {
#if HAVE_ASYNC_LDS
    __builtin_amdgcn_global_load_async_to_lds_b128(
        (__attribute__((address_space(1))) v4i_vs*)g,
        (__attribute__((address_space(3))) v4i_vs*)l,
        /*offset=*/0, /*cpol=*/0);
#else
    *(float4*)l = *(const float4*)g;
#endif
}

template <int N>
__device__ __forceinline__ void async_wait() {
#if HAVE_ASYNC_LDS
#  if HAVE_ASYNC_WAIT_BUILTIN
    __builtin_amdgcn_s_wait_asynccnt((unsigned short)N);
#  else
    asm volatile("s_wait_asynccnt %0" :: "i"(N) : "memory");
#  endif
#endif
}

// ---------------------------------------------------------------------------
// Kernel 1: depthwise causal conv (K=4) + bias + SiLU.  Memory bound.
// ---------------------------------------------------------------------------
__global__ void ssm_conv_silu(const float* __restrict__ x,
                              const float* __restrict__ conv_w,   // (D,1,K)
                              const float* __restrict__ conv_b,   // (D)
                              float* __restrict__ xc) {
    const int idx = blockIdx.x * blockDim.x + threadIdx.x;     // b*L*D + l*D + d
    const int d = idx & (DMODEL - 1);
    const int l = (idx >> 10) & (SEQ - 1);
    const float* w = conv_w + d * DCONV;
    float acc = conv_b[d];
#pragma unroll
    for (int j = 0; j < DCONV; ++j) {
        const int ls = l - (DCONV - 1) + j;
        const float xv = (ls >= 0) ? x[idx + (j - (DCONV - 1)) * DMODEL] : 0.0f;
        acc = fmaf(xv, w[j], acc);
    }
    xc[idx] = acc / (1.0f + __expf(-acc));   // SiLU
}

// ---------------------------------------------------------------------------
// Kernel 2: x_dbl = xc(16384x1024) @ x_proj_w^T(1024x96), fp32 WMMA 16x16x4.
// 8 waves/block own 8 consecutive M tiles and SHARE one 16-col B tile:
// B is staged global->LDS with async DMA, double buffered in K-chunks of 128,
// so the next chunk's DMA overlaps the current chunk's 32 WMMAs.
// LDS col stride 132 floats -> the 16x ds_load_b64 fragment reads are
// conflict-free across all 64 banks.
// ---------------------------------------------------------------------------
__global__ void ssm_proj_gemm(const float* __restrict__ xc,
                              const float* __restrict__ xpw,   // (96, 1024)
                              float* __restrict__ xdbl) {      // (16384, 96)
    __shared__ float bs[2][16 * BSTRIDE];

    const int lane    = threadIdx.x & 31;
    const int wave    = threadIdx.x >> 5;
    const int rowbase = (blockIdx.x * 8 + wave) * 16;
    const int colbase = blockIdx.y * 16;
    const int mn      = lane & 15;            // M row for A, N col for B
    const int kb      = (lane >> 4) << 1;     // lane-local K base (0 or 2)

    const float* arow = xc + (size_t)(rowbase + mn) * DMODEL + kb;

    // stage one 16 x KC chunk of B into LDS: 512 float4s, 2 per thread
    auto stage = [&](int kc, int buf) {
#pragma unroll
        for (int j = 0; j < 2; ++j) {
            const int f   = (int)threadIdx.x + 256 * j;   // float4 id 0..511
            const int col = f >> 5;                       // 0..15
            const int kk  = (f & 31) << 2;                // 0..124
            async_copy16(xpw + (size_t)(colbase + col) * DMODEL + kc + kk,
                         &bs[buf][col * BSTRIDE + kk]);
        }
    };

    stage(0, 0);
    v8f c = {};
    for (int ch = 0; ch < DMODEL / KC; ++ch) {
        if (ch < DMODEL / KC - 1) {           // overlap next DMA with compute
            stage((ch + 1) * KC, (ch + 1) & 1);
            async_wait<2>();                  // ops retire in order: chunk ch done
        } else {
            async_wait<0>();
        }
        __syncthreads();

        const int    buf  = ch & 1;
        const float* acol = arow + ch * KC;
        const float* bcol = &bs[buf][mn * BSTRIDE + kb];
#pragma unroll
        for (int k0 = 0; k0 < KC; k0 += 4) {
            v2f a = *(const v2f*)(acol + k0);   // global_load_b64
            v2f b = *(const v2f*)(bcol + k0);   // ds_load_b64, bank-conflict-free
            c = __builtin_amdgcn_wmma_f32_16x16x4_f32(
                    false, a, false, b, (short)0, c, false, false);
        }
        __syncthreads();                      // buf may be overwritten next iter
    }
#pragma unroll
    for (int v = 0; v < 8; ++v) {
        const int row = rowbase + v + ((lane >> 4) << 3);
        xdbl[(size_t)row * E_DIM + colbase + mn] = c[v];
    }
}

// ---------------------------------------------------------------------------
// Kernel 3: delta = softplus(x_dbl[:, :64] @ dt_proj_w^T + b).  K=64 -> one
// async-staged 4KB B tile shared by all 8 waves, then 16 WMMAs + fused
// bias/softplus epilogue.
// ---------------------------------------------------------------------------
__global__ void ssm_delta_gemm(const float* __restrict__ xdbl,  // (16384, 96)
                               const float* __restrict__ dtw,   // (1024, 64)
                               const float* __restrict__ dtb,   // (1024)
                               float* __restrict__ delta) {     // (16384, 1024)
    __shared__ float bs2[16 * BSTRIDE2];

    const int lane    = threadIdx.x & 31;
    const int wave    = threadIdx.x >> 5;
    const int rowbase = (blockIdx.x * 8 + wave) * 16;
    const int colbase = blockIdx.y * 16;
    const int mn      = lane & 15;
    const int kb      = (lane >> 4) << 1;

    {   // stage B tile: 16 cols x 64 k = 256 float4s, 1 per thread
        const int f   = (int)threadIdx.x;
        const int col = f >> 4;               // 0..15
        const int kk  = (f & 15) << 2;        // 0..60
        async_copy16(dtw + (size_t)(colbase + col) * DTRANK + kk,
                     &bs2[col * BSTRIDE2 + kk]);
        async_wait<0>();
        __syncthreads();
    }

    const float* arow = xdbl + (size_t)(rowbase + mn) * E_DIM + kb;
    const float* bcol = &bs2[mn * BSTRIDE2 + kb];

    v8f c = {};
#pragma unroll
    for (int k0 = 0; k0 < DTRANK; k0 += 4) {
        v2f a = *(const v2f*)(arow + k0);
        v2f b = *(const v2f*)(bcol + k0);
        c = __builtin_amdgcn_wmma_f32_16x16x4_f32(
                false, a, false, b, (short)0, c, false, false);
    }
    const int   dcol = colbase + mn;
    const float bias = dtb[dcol];
#pragma unroll
    for (int v = 0; v < 8; ++v) {
        const int row = rowbase + v + ((lane >> 4) << 3);
        const float val = c[v] + bias;
        const float sp  = (val > 20.0f) ? val : log1pf(__expf(val));
        delta[(size_t)row * DMODEL + dcol] = sp;
    }
}

// ---------------------------------------------------------------------------
// Kernel 4: selective scan.  One lane per (d, n): wave32 = 2 channels x 16
// states.  Serial over t; n-reduction = 4 shfl_xor rounds in 16-lane groups.
// A = -exp(A_log) hoisted; global_prefetch_b8 hides L2 latency on the chain.
// ---------------------------------------------------------------------------
__global__ void ssm_scan(const float* __restrict__ delta,  // (B,L,D)
                         const float* __restrict__ xc,     // (B,L,D)
                         const float* __restrict__ xdbl,   // (B*L,96): B@64, C@80
                         const float* __restrict__ A_log,  // (D, N)
                         const float* __restrict__ Dp,     // (D)
                         float* __restrict__ out) {        // (B,L,D)
    const int lane = threadIdx.x & 31;
    const int wave = threadIdx.x >> 5;
    const int b    = blockIdx.y;
    const int d    = blockIdx.x * 16 + wave * 2 + (lane >> 4);
    const int n    = lane & 15;

    const float A  = -__expf(A_log[d * DSTATE + n]);
    const float Dv = Dp[d];

    const float* dptr  = delta + (size_t)b * SEQ * DMODEL + d;
    const float* xptr  = xc    + (size_t)b * SEQ * DMODEL + d;
    const float* bcptr = xdbl  + (size_t)b * SEQ * E_DIM  + DTRANK + n;
    float*       optr  = out   + (size_t)b * SEQ * DMODEL + d;

    float h = 0.0f;
#pragma unroll 2
    for (int t = 0; t < SEQ; ++t) {
        if (t + 16 < SEQ) {
            __builtin_prefetch(dptr  + (t + 16) * DMODEL, 0, 0);
            __builtin_prefetch(xptr  + (t + 16) * DMODEL, 0, 0);
            __builtin_prefetch(bcptr + (size_t)(t + 16) * E_DIM, 0, 0);
        }
        const float dv = dptr[t * DMODEL];
        const float xv = xptr[t * DMODEL];
        const float Bv = bcptr[(size_t)t * E_DIM];
        const float Cv = bcptr[(size_t)t * E_DIM + DSTATE];

        const float dA = __expf(dv * A);
        h = fmaf(dA, h, dv * xv * Bv);

        float p = h * Cv;                  // reduce over n within 16 lanes
        p += __shfl_xor(p, 1, 16);
        p += __shfl_xor(p, 2, 16);
        p += __shfl_xor(p, 4, 16);
        p += __shfl_xor(p, 8, 16);
        if (n == 0) optr[t * DMODEL] = fmaf(Dv, xv, p);
    }
}

// ---------------------------------------------------------------------------
extern "C" void kernel_launch(void* const* d_in, const int* in_sizes, int n_in,
                              void* d_out, int out_size, void* d_ws, size_t ws_size,
                              hipStream_t stream) {
    const float* x      = (const float*)d_in[0];  // (8,2048,1024)
    const float* A_log  = (const float*)d_in[1];  // (1024,16)
    const float* Dp     = (const float*)d_in[2];  // (1024)
    const float* xpw    = (const float*)d_in[3];  // (96,1024)
    const float* dtw    = (const float*)d_in[4];  // (1024,64)
    const float* dtb    = (const float*)d_in[5];  // (1024)
    const float* convw  = (const float*)d_in[6];  // (1024,1,4)
    const float* convb  = (const float*)d_in[7];  // (1024)
    float*       out    = (float*)d_out;

    float* xc    = (float*)d_ws;                          // 64 MB
    float* delta = xc    + (size_t)MROWS * DMODEL;        // 64 MB
    float* xdbl  = delta + (size_t)MROWS * DMODEL;        // 6 MB

    ssm_conv_silu<<<(MROWS * DMODEL) / 256, 256, 0, stream>>>(x, convw, convb, xc);
    ssm_proj_gemm<<<dim3(MROWS / 16 / 8, E_DIM / 16), 256, 0, stream>>>(xc, xpw, xdbl);
    ssm_delta_gemm<<<dim3(MROWS / 16 / 8, DMODEL / 16), 256, 0, stream>>>(xdbl, dtw, dtb, delta);
    ssm_scan<<<dim3(DMODEL / 16, B_SZ), 256, 0, stream>>>(delta, xc, xdbl, A_log, Dp, out);
}